// Filtration_43482248905284
// MI455X (gfx1250) — compile-verified
//
#include <hip/hip_runtime.h>

typedef float v2f __attribute__((ext_vector_type(2)));
typedef float v8f __attribute__((ext_vector_type(8)));

__device__ __forceinline__ float lrelu(float x) { return x > 0.0f ? x : 0.01f * x; }

// D = A(16x4) * B(4x16) + C, fp32 exact.
// A layout: lane<16 holds A[m][0..1], lane>=16 holds A[m][2..3], m=lane&15
// B layout: lane<16 holds B[0..1][n], lane>=16 holds B[2..3][n], n=lane&15
__device__ __forceinline__ v8f wmma4(float ax, float ay, float bx, float by, v8f c) {
  v2f a; a.x = ax; a.y = ay;
  v2f b; b.x = bx; b.y = by;
  return __builtin_amdgcn_wmma_f32_16x16x4_f32(false, a, false, b, (short)0, c,
                                               false, false);
}

// ---------------------------------------------------------------- utilities
__global__ void zero_kernel(float* __restrict__ p, size_t total) {
  size_t i = ((size_t)blockIdx.x * blockDim.x + threadIdx.x) * 4;
  if (i < total) *(float4*)(p + i) = make_float4(0.f, 0.f, 0.f, 0.f);
}

// h0[node, 0:128] = concat(emb_deg[deg[node]], emb_lab[lab[node]]); 1 wave/node
__global__ void embed_kernel(const int* __restrict__ deg, const int* __restrict__ lab,
                             const float* __restrict__ ed, const float* __restrict__ el,
                             float* __restrict__ h0, int n) {
  int t = blockIdx.x * blockDim.x + threadIdx.x;
  int node = t >> 5, lane = t & 31;
  if (node >= n) return;
  float4 v;
  if (lane < 16) v = *(const float4*)(ed + (size_t)deg[node] * 64 + lane * 4);
  else           v = *(const float4*)(el + (size_t)lab[node] * 64 + (lane - 16) * 4);
  *(float4*)(h0 + (size_t)node * 128 + lane * 4) = v;
}

// agg[dst] += h[src] for every edge. (1<<lpeShift) lanes handle one edge,
// each lane owns a float4 slice -> coalesced gather + coalesced f32 atomics.
__global__ void agg_kernel(const float* __restrict__ h, const int* __restrict__ src,
                           const int* __restrict__ dst, float* __restrict__ agg,
                           int nEdges, int dimf, int lpeShift) {
  long t = (long)blockIdx.x * blockDim.x + threadIdx.x;
  long e = t >> lpeShift;
  if (e >= nEdges) return;
  int f = ((int)t & ((1 << lpeShift) - 1)) << 2;
  int s = src[e], d = dst[e];
  float4 v = *(const float4*)(h + (size_t)s * dimf + f);
  float* p = agg + (size_t)d * dimf + f;
  atomicAdd(p + 0, v.x);
  atomicAdd(p + 1, v.y);
  atomicAdd(p + 2, v.z);
  atomicAdd(p + 3, v.w);
}

// ------------------------------------------------- GIN MLP (WMMA), 4 waves/block
// y = lrelu((1+eps)*h + agg) @ w1 + b1) @ w2 + b2   (pre-BN), ODIM fixed 64.
// Also accumulates per-feature sum / sum-of-squares for BatchNorm.
template <int KDIM, int HDIM>
__global__ __launch_bounds__(128) void gin_mlp_kernel(
    const float* __restrict__ h, const float* __restrict__ agg,
    const float* __restrict__ eps_p,
    const float* __restrict__ w1, const float* __restrict__ b1,
    const float* __restrict__ w2, const float* __restrict__ b2,
    float* __restrict__ y, float* __restrict__ bnsum, float* __restrict__ bnsq,
    int ntiles) {
  constexpr int KPAD = KDIM + 2;   // odd dword-pair stride -> no bank conflicts
  constexpr int HT = HDIM / 16;
  __shared__ float s_x[4][16][KPAD];
  __shared__ float s_sum[64];
  __shared__ float s_sq[64];

  const int lane = threadIdx.x & 31;
  const int w = threadIdx.x >> 5;
  const int mrow = lane & 15;            // row for A, column for B/C
  const int khalf = (lane >> 4) << 1;    // 0 or 2
  const int rbase = (lane >> 4) << 3;    // C rows 0..7 or 8..15
  const int tile = blockIdx.x * 4 + w;
  const bool valid = tile < ntiles;      // wave-uniform -> EXEC all-1 for WMMA

  if (threadIdx.x < 64) { s_sum[threadIdx.x] = 0.f; s_sq[threadIdx.x] = 0.f; }

  // phase 1: x = (1+eps)*h + agg  -> LDS tile [16][KDIM]
  if (valid) {
    const float epsv = 1.0f + eps_p[0];
    const size_t base = (size_t)tile * 16 * KDIM;
    for (int idx = lane; idx < 16 * (KDIM / 4); idx += 32) {
      int r = idx / (KDIM / 4);
      int c = (idx % (KDIM / 4)) * 4;
      float4 hv = *(const float4*)(h + base + (size_t)r * KDIM + c);
      float4 av = *(const float4*)(agg + base + (size_t)r * KDIM + c);
      s_x[w][r][c + 0] = epsv * hv.x + av.x;
      s_x[w][r][c + 1] = epsv * hv.y + av.y;
      s_x[w][r][c + 2] = epsv * hv.z + av.z;
      s_x[w][r][c + 3] = epsv * hv.w + av.w;
    }
  }
  __syncthreads();

  // phase 2: hidden = x @ w1
  v8f acc[HT];
  {
    v8f z = {};
#pragma unroll
    for (int t = 0; t < HT; ++t) acc[t] = z;
  }
  if (valid) {
    for (int k = 0; k < KDIM; k += 4) {
      float ax = s_x[w][mrow][k + khalf];
      float ay = s_x[w][mrow][k + khalf + 1];
      const float* wp = w1 + (size_t)(k + khalf) * HDIM + mrow;
#pragma unroll
      for (int t = 0; t < HT; ++t)   // N-tile inner: no back-to-back RAW on acc
        acc[t] = wmma4(ax, ay, wp[t * 16], wp[HDIM + t * 16], acc[t]);
    }
  }
  __syncthreads();

  // phase 3: lrelu(hidden + b1) -> LDS (reuse tile buffer, HDIM <= KDIM)
  if (valid) {
#pragma unroll
    for (int t = 0; t < HT; ++t) {
      int col = t * 16 + mrow;
      float bb = b1[col];
#pragma unroll
      for (int v = 0; v < 8; ++v)
        s_x[w][rbase + v][col] = lrelu(acc[t][v] + bb);
    }
  }
  __syncthreads();

  // phase 4: y = hidden @ w2 + b2 ; accumulate BN statistics
  if (valid) {
    v8f acc2[4];
    {
      v8f z = {};
#pragma unroll
      for (int t = 0; t < 4; ++t) acc2[t] = z;
    }
    for (int k = 0; k < HDIM; k += 4) {
      float ax = s_x[w][mrow][k + khalf];
      float ay = s_x[w][mrow][k + khalf + 1];
      const float* wp = w2 + (size_t)(k + khalf) * 64 + mrow;
#pragma unroll
      for (int t = 0; t < 4; ++t)
        acc2[t] = wmma4(ax, ay, wp[t * 16], wp[64 + t * 16], acc2[t]);
    }
#pragma unroll
    for (int t = 0; t < 4; ++t) {
      int col = t * 16 + mrow;
      float bb = b2[col];
      float psum = 0.f, psq = 0.f;
#pragma unroll
      for (int v = 0; v < 8; ++v) {
        float val = acc2[t][v] + bb;
        y[((size_t)tile * 16 + rbase + v) * 64 + col] = val;
        psum += val;
        psq += val * val;
      }
      atomicAdd(&s_sum[col], psum);   // ds_add_f32
      atomicAdd(&s_sq[col], psq);
    }
  }
  __syncthreads();

  if (threadIdx.x < 64) {
    atomicAdd(bnsum + threadIdx.x, s_sum[threadIdx.x]);
    atomicAdd(bnsq + threadIdx.x, s_sq[threadIdx.x]);
  }
}

// -------------------------------------------- FC head GEMM (K=256), 2 waves/block
__global__ __launch_bounds__(64) void fc_kernel(
    const float* __restrict__ h0, const float* __restrict__ h1,
    const float* __restrict__ h2, const float* __restrict__ wf,
    const float* __restrict__ bf, float* __restrict__ y,
    float* __restrict__ bnsum, float* __restrict__ bnsq, int ntiles) {
  __shared__ float s_x[2][16][258];
  __shared__ float s_sum[64];
  __shared__ float s_sq[64];

  const int lane = threadIdx.x & 31;
  const int w = threadIdx.x >> 5;
  const int mrow = lane & 15;
  const int khalf = (lane >> 4) << 1;
  const int rbase = (lane >> 4) << 3;
  const int tile = blockIdx.x * 2 + w;
  const bool valid = tile < ntiles;

  if (threadIdx.x < 64) { s_sum[threadIdx.x] = 0.f; s_sq[threadIdx.x] = 0.f; }

  if (valid) {  // gather concat(h0,h1,h2) -> LDS tile [16][256]
    for (int idx = lane; idx < 16 * 64; idx += 32) {
      int r = idx >> 6;
      int g = idx & 63;
      int c = g * 4;
      size_t row = (size_t)tile * 16 + r;
      float4 v;
      if (g < 32)      v = *(const float4*)(h0 + row * 128 + c);
      else if (g < 48) v = *(const float4*)(h1 + row * 64 + (c - 128));
      else             v = *(const float4*)(h2 + row * 64 + (c - 192));
      s_x[w][r][c + 0] = v.x;
      s_x[w][r][c + 1] = v.y;
      s_x[w][r][c + 2] = v.z;
      s_x[w][r][c + 3] = v.w;
    }
  }
  __syncthreads();

  if (valid) {
    v8f acc[4];
    {
      v8f z = {};
#pragma unroll
      for (int t = 0; t < 4; ++t) acc[t] = z;
    }
    for (int k = 0; k < 256; k += 4) {
      float ax = s_x[w][mrow][k + khalf];
      float ay = s_x[w][mrow][k + khalf + 1];
      const float* wp = wf + (size_t)(k + khalf) * 64 + mrow;
#pragma unroll
      for (int t = 0; t < 4; ++t)
        acc[t] = wmma4(ax, ay, wp[t * 16], wp[64 + t * 16], acc[t]);
    }
#pragma unroll
    for (int t = 0; t < 4; ++t) {
      int col = t * 16 + mrow;
      float bb = bf[col];
      float psum = 0.f, psq = 0.f;
#pragma unroll
      for (int v = 0; v < 8; ++v) {
        float val = acc[t][v] + bb;
        y[((size_t)tile * 16 + rbase + v) * 64 + col] = val;
        psum += val;
        psq += val * val;
      }
      atomicAdd(&s_sum[col], psum);
      atomicAdd(&s_sq[col], psq);
    }
  }
  __syncthreads();

  if (threadIdx.x < 64) {
    atomicAdd(bnsum + threadIdx.x, s_sum[threadIdx.x]);
    atomicAdd(bnsq + threadIdx.x, s_sq[threadIdx.x]);
  }
}

// ------------------------------------------------------------- BatchNorm pieces
__global__ void bn_finalize_kernel(const float* __restrict__ sum,
                                   const float* __restrict__ sq,
                                   const float* __restrict__ g,
                                   const float* __restrict__ b, float inv_n,
                                   float* __restrict__ sc, float* __restrict__ sh) {
  int f = threadIdx.x;  // 64 threads
  float mean = sum[f] * inv_n;
  float var = sq[f] * inv_n - mean * mean;  // biased var, matches jnp.var
  float scale = g[f] * rsqrtf(var + 1e-5f);
  sc[f] = scale;
  sh[f] = b[f] - mean * scale;
}

__global__ void bn_apply_kernel(const float* __restrict__ y,
                                const float* __restrict__ sc,
                                const float* __restrict__ sh,
                                float* __restrict__ h, size_t total) {
  size_t i = ((size_t)blockIdx.x * blockDim.x + threadIdx.x) * 4;
  if (i >= total) return;
  int f = (int)(i & 63);
  float4 v = *(const float4*)(y + i);
  float4 s = *(const float4*)(sc + f);
  float4 t = *(const float4*)(sh + f);
  float4 o;
  o.x = lrelu(v.x * s.x + t.x);
  o.y = lrelu(v.y * s.y + t.y);
  o.z = lrelu(v.z * s.z + t.z);
  o.w = lrelu(v.w * s.w + t.w);
  *(float4*)(h + i) = o;
}

__global__ void final_kernel(const float* __restrict__ y,
                             const float* __restrict__ sc,
                             const float* __restrict__ sh,
                             const float* __restrict__ w2,
                             const float* __restrict__ b2,
                             float* __restrict__ out, int n) {
  int i = blockIdx.x * blockDim.x + threadIdx.x;
  if (i >= n) return;
  float acc = b2[0];
  const float* yr = y + (size_t)i * 64;
#pragma unroll
  for (int f = 0; f < 64; f += 4) {
    float4 v = *(const float4*)(yr + f);
    float4 s = *(const float4*)(sc + f);
    float4 t = *(const float4*)(sh + f);
    float4 wv = *(const float4*)(w2 + f);
    acc += lrelu(v.x * s.x + t.x) * wv.x;
    acc += lrelu(v.y * s.y + t.y) * wv.y;
    acc += lrelu(v.z * s.z + t.z) * wv.z;
    acc += lrelu(v.w * s.w + t.w) * wv.w;
  }
  out[i] = 1.0f / (1.0f + expf(-acc));
}

// ------------------------------------------------------------------- launcher
extern "C" void kernel_launch(void* const* d_in, const int* in_sizes, int n_in,
                              void* d_out, int out_size, void* d_ws, size_t ws_size,
                              hipStream_t stream) {
  const int* node_deg = (const int*)d_in[0];
  const int* node_lab = (const int*)d_in[1];
  const int* edge_index = (const int*)d_in[2];
  const float* emb_deg = (const float*)d_in[3];
  const float* emb_lab = (const float*)d_in[4];
  const float* l0_w1 = (const float*)d_in[5];
  const float* l0_b1 = (const float*)d_in[6];
  const float* l0_w2 = (const float*)d_in[7];
  const float* l0_b2 = (const float*)d_in[8];
  const float* l0_eps = (const float*)d_in[9];
  const float* bn0_g = (const float*)d_in[10];
  const float* bn0_b = (const float*)d_in[11];
  const float* l1_w1 = (const float*)d_in[12];
  const float* l1_b1 = (const float*)d_in[13];
  const float* l1_w2 = (const float*)d_in[14];
  const float* l1_b2 = (const float*)d_in[15];
  const float* l1_eps = (const float*)d_in[16];
  const float* bn1_g = (const float*)d_in[17];
  const float* bn1_b = (const float*)d_in[18];
  const float* fc_w1 = (const float*)d_in[19];
  const float* fc_b1 = (const float*)d_in[20];
  const float* fc_bn_g = (const float*)d_in[21];
  const float* fc_bn_b = (const float*)d_in[22];
  const float* fc_w2 = (const float*)d_in[23];
  const float* fc_b2 = (const float*)d_in[24];

  const int N = in_sizes[0];
  const int E = in_sizes[2] / 2;
  const int* src = edge_index;
  const int* dst = edge_index + E;
  const size_t Nl = (size_t)N;

  // workspace layout (floats): needs (N*448 + 256)*4 bytes (~179 MB @ N=100k)
  float* ws = (float*)d_ws;
  float* h0 = ws;                 // N*128
  float* h1 = h0 + Nl * 128;      // N*64
  float* h2 = h1 + Nl * 64;       // N*64
  float* agg = h2 + Nl * 64;      // N*128 (reused for both layers)
  float* ybuf = agg + Nl * 128;   // N*64  (y1 -> y2 -> y3, sequential reuse)
  float* bsum = ybuf + Nl * 64;   // 64
  float* bsq = bsum + 64;         // 64
  float* bsc = bsq + 64;          // 64
  float* bsh = bsc + 64;          // 64

  const float invN = 1.0f / (float)N;
  const int ntiles = (N + 15) / 16;
  auto cdiv = [](long a, long b) { return (int)((a + b - 1) / b); };

  // h0 embedding
  embed_kernel<<<cdiv((long)N * 32, 256), 256, 0, stream>>>(
      node_deg, node_lab, emb_deg, emb_lab, h0, N);

  // ---- GIN layer 0 (K=128 -> H=128 -> 64)
  zero_kernel<<<cdiv((long)Nl * 128 / 4, 256), 256, 0, stream>>>(agg, Nl * 128);
  zero_kernel<<<1, 32, 0, stream>>>(bsum, 128);
  agg_kernel<<<cdiv((long)E * 32, 256), 256, 0, stream>>>(h0, src, dst, agg, E, 128, 5);
  gin_mlp_kernel<128, 128><<<cdiv(ntiles, 4), 128, 0, stream>>>(
      h0, agg, l0_eps, l0_w1, l0_b1, l0_w2, l0_b2, ybuf, bsum, bsq, ntiles);
  bn_finalize_kernel<<<1, 64, 0, stream>>>(bsum, bsq, bn0_g, bn0_b, invN, bsc, bsh);
  bn_apply_kernel<<<cdiv((long)Nl * 64 / 4, 256), 256, 0, stream>>>(
      ybuf, bsc, bsh, h1, Nl * 64);

  // ---- GIN layer 1 (K=64 -> H=64 -> 64)
  zero_kernel<<<cdiv((long)Nl * 64 / 4, 256), 256, 0, stream>>>(agg, Nl * 64);
  zero_kernel<<<1, 32, 0, stream>>>(bsum, 128);
  agg_kernel<<<cdiv((long)E * 16, 256), 256, 0, stream>>>(h1, src, dst, agg, E, 64, 4);
  gin_mlp_kernel<64, 64><<<cdiv(ntiles, 4), 128, 0, stream>>>(
      h1, agg, l1_eps, l1_w1, l1_b1, l1_w2, l1_b2, ybuf, bsum, bsq, ntiles);
  bn_finalize_kernel<<<1, 64, 0, stream>>>(bsum, bsq, bn1_g, bn1_b, invN, bsc, bsh);
  bn_apply_kernel<<<cdiv((long)Nl * 64 / 4, 256), 256, 0, stream>>>(
      ybuf, bsc, bsh, h2, Nl * 64);

  // ---- FC head
  zero_kernel<<<1, 32, 0, stream>>>(bsum, 128);
  fc_kernel<<<cdiv(ntiles, 2), 64, 0, stream>>>(h0, h1, h2, fc_w1, fc_b1, ybuf,
                                                bsum, bsq, ntiles);
  bn_finalize_kernel<<<1, 64, 0, stream>>>(bsum, bsq, fc_bn_g, fc_bn_b, invN, bsc, bsh);
  final_kernel<<<cdiv(N, 256), 256, 0, stream>>>(ybuf, bsc, bsh, fc_w2, fc_b2,
                                                 (float*)d_out, N);
}